// Code2Vec_5798205849679
// MI455X (gfx1250) — compile-verified
//
#include <hip/hip_runtime.h>
#include <hip/hip_bf16.h>

typedef __attribute__((ext_vector_type(2))) float v2f;
typedef __attribute__((ext_vector_type(8))) float v8f;

namespace {
constexpr int kB    = 64;
constexpr int kT    = 512;
constexpr int kE    = 128;
constexpr int kK    = 384;   // 3*E
constexpr int kOUT  = 1000;
constexpr int kPAD  = 1;

constexpr int kKC        = 32;   // K-chunk staged in LDS
constexpr int kLdsStride = 34;   // padded row stride (floats), even for b64 align
}

// ---------------------------------------------------------------------------
// Kernel 1: fused gather + f32 WMMA GEMM (c @ W^T) + tanh + (x . a)
// grid: (B*T)/64 blocks of 128 threads (4 waves); wave = one 16-row M tile.
// ---------------------------------------------------------------------------
__global__ __launch_bounds__(128)
void k1_gemm_tanh(const int* __restrict__ starts, const int* __restrict__ paths,
                  const int* __restrict__ ends,
                  const float* __restrict__ node_emb,
                  const float* __restrict__ path_emb,
                  const float* __restrict__ W, const float* __restrict__ a,
                  float* __restrict__ x_out, float* __restrict__ s_out)
{
    __shared__ float wlds[kE * kLdsStride];   // 128 x 34 floats = 17408 B

    const int tid  = threadIdx.x;
    const int wave = tid >> 5;
    const int lane = tid & 31;
    const int half = lane >> 4;    // 0: K pair {0,1}, 1: K pair {2,3}
    const int l16  = lane & 15;

    const int row_base = blockIdx.x * 64 + wave * 16;   // token tile base
    const int tok      = row_base + l16;                // A-matrix row (m = lane%16)

    // Per-lane gather bases for the three concat segments of c
    const int si = starts[tok];
    const int pi = paths[tok];
    const int ei = ends[tok];
    const float* base0 = node_emb + (size_t)si * kE;
    const float* base1 = path_emb + (size_t)pi * kE;
    const float* base2 = node_emb + (size_t)ei * kE;
    const float m0 = (si == kPAD) ? 0.0f : 1.0f;   // padding_idx=1 -> zero row
    const float m2 = (ei == kPAD) ? 0.0f : 1.0f;

    v8f acc[8] = {};   // 8 N-tiles of 16x16 f32 accumulators

    for (int kc = 0; kc < kK; kc += kKC) {
        __syncthreads();
        // Cooperatively stage W[n][kc..kc+31] (B^T slice) into LDS
        {
            const int kl = tid & 31;
            const int n0 = tid >> 5;           // 0..3
            #pragma unroll
            for (int n = n0; n < kE; n += 4) {
                wlds[n * kLdsStride + kl] = W[(size_t)n * kK + kc + kl];
            }
        }
        __syncthreads();

        #pragma unroll
        for (int ks = 0; ks < kKC; ks += 4) {
            // A fragment: 2 consecutive K values per lane (gathered from c)
            const int klane = ks + half * 2;           // k within chunk
            const int kg    = kc + klane;              // global k in [0,384)
            const int seg   = kg >> 7;
            const int ko    = kg & 127;
            const float* p  = (seg == 0) ? base0 : ((seg == 1) ? base1 : base2);
            const float msk = (seg == 0) ? m0   : ((seg == 1) ? 1.0f  : m2);
            v2f afrag = *(const v2f*)(p + ko);
            afrag *= msk;

            // 8 N-tiles: B fragment from LDS, then f32 WMMA
            #pragma unroll
            for (int nt = 0; nt < 8; ++nt) {
                const float* lp = &wlds[(nt * 16 + l16) * kLdsStride + klane];
                v2f bfrag = *(const v2f*)lp;
                acc[nt] = __builtin_amdgcn_wmma_f32_16x16x4_f32(
                    false, afrag, false, bfrag, (short)0, acc[nt], false, false);
            }
        }
    }

    // Epilogue: tanh, store x, fold in s = x . a
    float afr[8];
    #pragma unroll
    for (int nt = 0; nt < 8; ++nt) afr[nt] = a[nt * 16 + l16];

    float prt[8];
    #pragma unroll
    for (int j = 0; j < 8; ++j) prt[j] = 0.0f;

    #pragma unroll
    for (int nt = 0; nt < 8; ++nt) {
        #pragma unroll
        for (int j = 0; j < 8; ++j) {
            const float xv = tanhf(acc[nt][j]);
            const int m = j + half * 8;                       // row within tile
            x_out[(size_t)(row_base + m) * kE + nt * 16 + l16] = xv;
            prt[j] += xv * afr[nt];
        }
    }

    // Butterfly reduce across each 16-lane half (n dimension)
    #pragma unroll
    for (int j = 0; j < 8; ++j) {
        float v = prt[j];
        v += __shfl_xor(v, 1, 32);
        v += __shfl_xor(v, 2, 32);
        v += __shfl_xor(v, 4, 32);
        v += __shfl_xor(v, 8, 32);
        prt[j] = v;
    }
    if (l16 == 0) {
        #pragma unroll
        for (int j = 0; j < 8; ++j) {
            s_out[row_base + half * 8 + j] = prt[j];
        }
    }
}

// ---------------------------------------------------------------------------
// Kernel 2: per-batch softmax over T, then mask t < length[b] (no renorm)
// grid: B blocks of T=512 threads (16 waves)
// ---------------------------------------------------------------------------
__global__ __launch_bounds__(512)
void k2_softmax_mask(const float* __restrict__ s, const int* __restrict__ length,
                     float* __restrict__ w)
{
    __shared__ float red[16];
    __shared__ float bmax, bsum;

    const int b  = blockIdx.x;
    const int t  = threadIdx.x;
    const int wv = t >> 5;
    const int ln = t & 31;

    const float val = s[b * kT + t];

    // block max
    float m = val;
    m = fmaxf(m, __shfl_xor(m, 1, 32));
    m = fmaxf(m, __shfl_xor(m, 2, 32));
    m = fmaxf(m, __shfl_xor(m, 4, 32));
    m = fmaxf(m, __shfl_xor(m, 8, 32));
    m = fmaxf(m, __shfl_xor(m, 16, 32));
    if (ln == 0) red[wv] = m;
    __syncthreads();
    if (t == 0) {
        float mm = red[0];
        for (int i = 1; i < 16; ++i) mm = fmaxf(mm, red[i]);
        bmax = mm;
    }
    __syncthreads();

    const float e = __expf(val - bmax);

    // block sum
    float sm = e;
    sm += __shfl_xor(sm, 1, 32);
    sm += __shfl_xor(sm, 2, 32);
    sm += __shfl_xor(sm, 4, 32);
    sm += __shfl_xor(sm, 8, 32);
    sm += __shfl_xor(sm, 16, 32);
    if (ln == 0) red[wv] = sm;
    __syncthreads();
    if (t == 0) {
        float s2 = 0.0f;
        for (int i = 0; i < 16; ++i) s2 += red[i];
        bsum = s2;
    }
    __syncthreads();

    const float msk = (t < length[b]) ? 1.0f : 0.0f;
    w[b * kT + t] = (e / bsum) * msk;
}

// ---------------------------------------------------------------------------
// Kernel 3: v[b,e] = sum_t x[b,t,e] * w[b,t]
// grid: B blocks of E=128 threads
// ---------------------------------------------------------------------------
__global__ __launch_bounds__(128)
void k3_pool(const float* __restrict__ x, const float* __restrict__ w,
             float* __restrict__ v)
{
    const int b = blockIdx.x;
    const int e = threadIdx.x;
    const float* xb = x + (size_t)b * kT * kE;
    const float* wb = w + b * kT;
    float acc = 0.0f;
    #pragma unroll 4
    for (int t = 0; t < kT; ++t) {
        acc += xb[(size_t)t * kE + e] * wb[t];
    }
    v[b * kE + e] = acc;
}

// ---------------------------------------------------------------------------
// Kernel 4: out[b,o] = v[b,:] . out_W[o,:] + out_b[o]
// grid: B blocks of 256 threads, v cached in LDS, float4 dots
// ---------------------------------------------------------------------------
__global__ __launch_bounds__(256)
void k4_out(const float* __restrict__ v, const float* __restrict__ out_W,
            const float* __restrict__ out_b, float* __restrict__ out)
{
    __shared__ float vb[kE];
    const int b = blockIdx.x;
    if (threadIdx.x < kE) vb[threadIdx.x] = v[b * kE + threadIdx.x];
    __syncthreads();

    for (int o = threadIdx.x; o < kOUT; o += 256) {
        const float4* wr = (const float4*)(out_W + (size_t)o * kE);
        const float4* vv = (const float4*)vb;
        float acc = out_b[o];
        #pragma unroll
        for (int e4 = 0; e4 < kE / 4; ++e4) {
            const float4 wq = wr[e4];
            const float4 vq = vv[e4];
            acc += wq.x * vq.x + wq.y * vq.y + wq.z * vq.z + wq.w * vq.w;
        }
        out[b * kOUT + o] = acc;
    }
}

// ---------------------------------------------------------------------------
extern "C" void kernel_launch(void* const* d_in, const int* in_sizes, int n_in,
                              void* d_out, int out_size, void* d_ws, size_t ws_size,
                              hipStream_t stream)
{
    const int*   starts   = (const int*)d_in[0];
    const int*   paths    = (const int*)d_in[1];
    const int*   ends     = (const int*)d_in[2];
    const int*   length   = (const int*)d_in[3];
    const float* node_emb = (const float*)d_in[4];
    const float* path_emb = (const float*)d_in[5];
    const float* W        = (const float*)d_in[6];
    const float* a        = (const float*)d_in[7];
    const float* out_W    = (const float*)d_in[8];
    const float* out_b    = (const float*)d_in[9];
    float*       out      = (float*)d_out;

    // workspace layout (floats): x[B*T*E] | s[B*T] | w[B*T] | v[B*E]
    float* ws = (float*)d_ws;
    float* x  = ws;
    float* s  = x + (size_t)kB * kT * kE;
    float* wz = s + (size_t)kB * kT;
    float* v  = wz + (size_t)kB * kT;

    (void)in_sizes; (void)n_in; (void)out_size; (void)ws_size;

    k1_gemm_tanh<<<(kB * kT) / 64, 128, 0, stream>>>(
        starts, paths, ends, node_emb, path_emb, W, a, x, s);
    k2_softmax_mask<<<kB, kT, 0, stream>>>(s, length, wz);
    k3_pool<<<kB, kE, 0, stream>>>(x, wz, v);
    k4_out<<<kB, 256, 0, stream>>>(v, out_W, out_b, out);
}